// ChamferkNNDist_27015344292566
// MI455X (gfx1250) — compile-verified
//
#include <hip/hip_runtime.h>
#include <hip/hip_bf16.h>
#include <math.h>

typedef __attribute__((ext_vector_type(2))) float v2f;
typedef __attribute__((ext_vector_type(8))) float v8f;

#define BATCH 8
#define NPTS  4096
#define NTILE (NPTS / 16)   // 256 tiles of 16 points
#define FBIG  3.0e38f

// Branchless insert into ascending sorted 6-list of smallest values.
__device__ __forceinline__ void insert6(float (&s)[6], float v) {
    float t = v;
#pragma unroll
    for (int r = 0; r < 6; ++r) {
        float lo = fminf(s[r], t);
        float hi = fmaxf(s[r], t);
        s[r] = lo;
        t = hi;
    }
}

// ---------------------------------------------------------------------------
// Kernel 1: chamfer column minima.
// One wave per 16-wide tile of adv columns; sweep all ori row tiles via WMMA.
// D[i][j] = -2*<ori_i, adv_j> + ||ori_i||^2  (then + ||adv_j||^2 at the end)
// ---------------------------------------------------------------------------
__global__ __launch_bounds__(256) void chamfer_min_kernel(
    const float* __restrict__ adv, const float* __restrict__ ori,
    float* __restrict__ cmin) {
    const int wave = threadIdx.x >> 5;
    const int lane = threadIdx.x & 31;
    const int tile = blockIdx.x * 8 + wave;   // global j-tile, 0..B*NTILE-1
    const int b    = tile >> 8;               // NTILE == 256
    const int jt   = tile & (NTILE - 1);
    const int lp   = lane & 15;
    const int half = lane >> 4;

    // --- B matrix tile: adv points (columns), K-slot 3 = 1.0 ---
    const float* ap = adv + ((size_t)b * NPTS + (size_t)(jt * 16 + lp)) * 3;
    const float bx = ap[0], by = ap[1], bz = ap[2];
    const float ry = bx * bx + by * by + bz * bz;
    v2f Bm;
    Bm.x = half ? bz : bx;      // K=0 (lo half) / K=2 (hi half)
    Bm.y = half ? 1.0f : by;    // K=1 (lo half) / K=3 (hi half)

    const float* obase = ori + (size_t)b * NPTS * 3;
    float m = FBIG;

    for (int it = 0; it < NTILE; ++it) {
        const float* op = obase + (size_t)(it * 16 + lp) * 3;
        const float ox = op[0], oy = op[1], oz = op[2];
        const float rx = ox * ox + oy * oy + oz * oz;
        v2f Am;
        Am.x = half ? (-2.0f * oz) : (-2.0f * ox);
        Am.y = half ? rx : (-2.0f * oy);

        v8f c = {0.f, 0.f, 0.f, 0.f, 0.f, 0.f, 0.f, 0.f};
        v8f d = __builtin_amdgcn_wmma_f32_16x16x4_f32(
            false, Am, false, Bm, (short)0, c, false, false);

        // All 8 accumulator rows belong to this lane's column j -> fold min.
        float m0 = fminf(fminf(d[0], d[1]), fminf(d[2], d[3]));
        float m1 = fminf(fminf(d[4], d[5]), fminf(d[6], d[7]));
        m = fminf(m, fminf(m0, m1));
    }

    // Combine the two lane-halves (rows 0-7 vs 8-15 of each tile).
    m = fminf(m, __shfl_xor(m, 16, 32));
    m += ry;                                   // + ||adv_j||^2

    if (half == 0)
        cmin[(size_t)b * NPTS + (size_t)(jt * 16 + lp)] = m;
}

// ---------------------------------------------------------------------------
// Kernel 2: kNN average distance per adv point (drop self, keep 5 nearest).
// Same WMMA tiling over adv x adv; per-column sorted top-6 selection.
// ---------------------------------------------------------------------------
__global__ __launch_bounds__(256) void knn_value_kernel(
    const float* __restrict__ adv, float* __restrict__ value) {
    const int wave = threadIdx.x >> 5;
    const int lane = threadIdx.x & 31;
    const int tile = blockIdx.x * 8 + wave;
    const int b    = tile >> 8;
    const int jt   = tile & (NTILE - 1);
    const int lp   = lane & 15;
    const int half = lane >> 4;

    const float* abase = adv + (size_t)b * NPTS * 3;

    const float* ap = abase + (size_t)(jt * 16 + lp) * 3;
    const float bx = ap[0], by = ap[1], bz = ap[2];
    const float ry = bx * bx + by * by + bz * bz;
    v2f Bm;
    Bm.x = half ? bz : bx;
    Bm.y = half ? 1.0f : by;

    float s[6] = {FBIG, FBIG, FBIG, FBIG, FBIG, FBIG};

    for (int it = 0; it < NTILE; ++it) {
        const float* op = abase + (size_t)(it * 16 + lp) * 3;
        const float ox = op[0], oy = op[1], oz = op[2];
        const float rx = ox * ox + oy * oy + oz * oz;
        v2f Am;
        Am.x = half ? (-2.0f * oz) : (-2.0f * ox);
        Am.y = half ? rx : (-2.0f * oy);

        v8f c = {0.f, 0.f, 0.f, 0.f, 0.f, 0.f, 0.f, 0.f};
        v8f d = __builtin_amdgcn_wmma_f32_16x16x4_f32(
            false, Am, false, Bm, (short)0, c, false, false);

#pragma unroll
        for (int r = 0; r < 8; ++r) insert6(s, d[r]);
    }

    // Merge the two half-lists for this column (read first, then insert).
    float t[6];
#pragma unroll
    for (int r = 0; r < 6; ++r) t[r] = __shfl_xor(s[r], 16, 32);
#pragma unroll
    for (int r = 0; r < 6; ++r) insert6(s, t[r]);

    // Add ||adv_j||^2, clamp like reference (max with -1e-6), drop self (s[0]).
    float acc = 0.0f;
#pragma unroll
    for (int r = 1; r < 6; ++r) acc += fmaxf(s[r] + ry, -1e-6f);

    if (half == 0)
        value[(size_t)b * NPTS + (size_t)(jt * 16 + lp)] = acc * 0.2f;
}

// ---------------------------------------------------------------------------
// Kernel 3: deterministic reductions -> final scalar.
// ---------------------------------------------------------------------------
__device__ float blockReduceSum(float v, float* red) {
    const int t = threadIdx.x;
    red[t] = v;
    __syncthreads();
    for (int s = 128; s > 0; s >>= 1) {
        if (t < s) red[t] += red[t + s];
        __syncthreads();
    }
    float r = red[0];
    __syncthreads();
    return r;
}

__global__ __launch_bounds__(256) void finalize_kernel(
    const float* __restrict__ ws, float* __restrict__ out) {
    __shared__ float red[256];
    const int t = threadIdx.x;
    float ch_acc = 0.0f, knn_acc = 0.0f;

    for (int b = 0; b < BATCH; ++b) {
        const float* cm  = ws + (size_t)b * NPTS;
        const float* val = ws + (size_t)BATCH * NPTS + (size_t)b * NPTS;

        float s_min = 0.f, s_v = 0.f, s_v2 = 0.f;
        for (int j = t; j < NPTS; j += 256) {
            s_min += cm[j];
            const float v = val[j];
            s_v  += v;
            s_v2 += v * v;
        }
        s_min = blockReduceSum(s_min, red);
        s_v   = blockReduceSum(s_v, red);
        s_v2  = blockReduceSum(s_v2, red);

        const float mean = s_v / (float)NPTS;
        const float var  = (s_v2 - (float)NPTS * mean * mean) / (float)(NPTS - 1);
        const float stdv = sqrtf(fmaxf(var, 0.0f));
        const float thr  = mean + 1.05f * stdv;

        float s_mask = 0.f;
        for (int j = t; j < NPTS; j += 256) {
            const float v = val[j];
            s_mask += (v > thr) ? v : 0.0f;
        }
        s_mask = blockReduceSum(s_mask, red);

        ch_acc  += s_min  / (float)NPTS;
        knn_acc += s_mask / (float)NPTS;
    }

    if (t == 0)
        out[0] = 5.0f * (ch_acc / (float)BATCH) + 3.0f * (knn_acc / (float)BATCH);
}

// ---------------------------------------------------------------------------
extern "C" void kernel_launch(void* const* d_in, const int* in_sizes, int n_in,
                              void* d_out, int out_size, void* d_ws, size_t ws_size,
                              hipStream_t stream) {
    const float* adv = (const float*)d_in[0];   // [B, N, 3] f32
    const float* ori = (const float*)d_in[1];   // [B, N, 3] f32
    float* out = (float*)d_out;
    float* ws  = (float*)d_ws;                  // needs 2*B*N floats = 256 KB

    // 2048 j-tiles total, 8 waves (256 threads) per block -> 256 blocks.
    chamfer_min_kernel<<<256, 256, 0, stream>>>(adv, ori, ws);
    knn_value_kernel<<<256, 256, 0, stream>>>(adv, ws + (size_t)BATCH * NPTS);
    finalize_kernel<<<1, 256, 0, stream>>>(ws, out);
}